// SoftPositionalConstraint_47614007443700
// MI455X (gfx1250) — compile-verified
//
#include <hip/hip_runtime.h>
#include <math.h>

// Problem constants (from the reference)
#define NCONST 2097152               // constraints per side
#define NBODY  4194304               // bodies; == 2*NCONST
#define NEND   (2 * NCONST)          // concatenated endpoints
#define TPB    256
#define EPT    8                     // endpoints per thread
#define NBLK   (NEND / (TPB * EPT))  // 2048 blocks
#define DT_F   0.01f

#define GAS __attribute__((address_space(1)))
#define LAS __attribute__((address_space(3)))

typedef int v4i __attribute__((vector_size(4 * sizeof(int))));

#if defined(__has_builtin)
#  if __has_builtin(__builtin_amdgcn_global_load_async_to_lds_b128)
#    define HAVE_ALD 1
#  endif
#  if __has_builtin(__builtin_amdgcn_global_store_async_from_lds_b128)
#    define HAVE_AST 1
#  endif
#endif
#ifndef HAVE_ALD
#  define HAVE_ALD 0
#endif
#ifndef HAVE_AST
#  define HAVE_AST 0
#endif

// ---------------------------------------------------------------------------
// K1: gather + rotate, block-level mass-weighted reduction. The velocity->out
//     copy and the index staging both ride the async (ASYNCcnt) engine:
//     global->LDS loads, one s_wait_asynccnt, LDS->global stores, while the
//     VALU does the gathers / sincos / reduction.
// ---------------------------------------------------------------------------
__global__ __launch_bounds__(TPB) void k1_reduce_copy(
    const int* __restrict__ from_b, const int* __restrict__ to_b,
    const float2* __restrict__ from_rp, const float2* __restrict__ to_rp,
    const float2* __restrict__ pos, const float* __restrict__ ang,
    const float* __restrict__ mass, const float4* __restrict__ vel4,
    float4* __restrict__ out4, float4* __restrict__ partials)
{
    __shared__ int   lds_idx[TPB * EPT];          // 8 KB index tile
    __shared__ int   lds_vel[TPB * 16];           // 16 KB velocity tile
    __shared__ float swx[TPB / 32], swy[TPB / 32], swm[TPB / 32];

    const int tid = threadIdx.x;
    const int blk = blockIdx.x;
    const int e0  = blk * (TPB * EPT);            // first endpoint of block
    // NCONST is a multiple of the tile size, so a block never straddles halves
    const int* __restrict__ src =
        (e0 < NCONST) ? (from_b + e0) : (to_b + (e0 - NCONST));
    const int b4 = (e0 >> 1) + tid * 4;           // float4 = 2 bodies

#if HAVE_ALD
    // Async-stage this block's 2048 indices into LDS (ASYNCcnt path).
    {
        const int* gp = src + tid * 8;
        __builtin_amdgcn_global_load_async_to_lds_b128(
            (GAS v4i*)gp, (LAS v4i*)&lds_idx[tid * 8], 0, 0);
        __builtin_amdgcn_global_load_async_to_lds_b128(
            (GAS v4i*)(gp + 4), (LAS v4i*)&lds_idx[tid * 8 + 4], 0, 0);
    }
#endif

#if HAVE_ALD && HAVE_AST
    // Velocity -> out copy rides the async engine: global -> LDS now,
    // LDS -> global after the single asynccnt wait below. No VGPR traffic.
#pragma unroll
    for (int j = 0; j < 4; ++j) {
        __builtin_amdgcn_global_load_async_to_lds_b128(
            (GAS v4i*)(vel4 + b4 + j), (LAS v4i*)&lds_vel[(tid * 4 + j) * 4], 0, 0);
    }
#else
    {
        float4 c0 = vel4[b4 + 0];
        float4 c1 = vel4[b4 + 1];
        float4 c2 = vel4[b4 + 2];
        float4 c3 = vel4[b4 + 3];
        out4[b4 + 0] = c0;
        out4[b4 + 1] = c1;
        out4[b4 + 2] = c2;
        out4[b4 + 3] = c3;
    }
#endif

#if HAVE_ALD
    asm volatile("s_wait_asynccnt 0x0" ::: "memory");
#endif

#if HAVE_ALD && HAVE_AST
    // Kick the second half of the copy; completion is drained by S_ENDPGM's
    // implicit wait-idle, fully overlapped with the gather math below.
#pragma unroll
    for (int j = 0; j < 4; ++j) {
        __builtin_amdgcn_global_store_async_from_lds_b128(
            (GAS v4i*)(out4 + b4 + j), (LAS v4i*)&lds_vel[(tid * 4 + j) * 4], 0, 0);
    }
#endif

    int4 ia, ib;
#if HAVE_ALD
    ia = *(const int4*)&lds_idx[tid * 8];
    ib = *(const int4*)&lds_idx[tid * 8 + 4];
#else
    ia = *(const int4*)(src + tid * 8);
    ib = *(const int4*)(src + tid * 8 + 4);
#endif

    float wx = 0.f, wy = 0.f, wm = 0.f;
    const int idxs[8] = {ia.x, ia.y, ia.z, ia.w, ib.x, ib.y, ib.z, ib.w};
#pragma unroll
    for (int i = 0; i < 8; ++i) {
        const int idx = idxs[i];
        const float2 p = pos[idx];
        const float  a = ang[idx];
        const float  m = mass[idx];
        // rel_positions is concat(from_pos, to_pos) indexed by the BODY index
        const float2* rp = (idx < NCONST) ? from_rp : to_rp;
        const float2 r = rp[idx & (NCONST - 1)];
        float sa, ca;
        __sincosf(a, &sa, &ca);
        const float c = sa;        // cos(a - pi/2) =  sin a
        const float s = -ca;       // sin(a - pi/2) = -cos a
        const float ax = fmaf(c, r.x, fmaf(-s, r.y, p.x));
        const float ay = fmaf(s, r.x, fmaf( c, r.y, p.y));
        wx = fmaf(m, ax, wx);
        wy = fmaf(m, ay, wy);
        wm += m;
    }

    // wave32 reduction
#pragma unroll
    for (int o = 16; o; o >>= 1) {
        wx += __shfl_xor(wx, o, 32);
        wy += __shfl_xor(wy, o, 32);
        wm += __shfl_xor(wm, o, 32);
    }
    const int w = tid >> 5, l = tid & 31;
    if (l == 0) { swx[w] = wx; swy[w] = wy; swm[w] = wm; }
    __syncthreads();
    if (tid < TPB / 32) {          // 8 lanes of wave 0
        wx = swx[tid]; wy = swy[tid]; wm = swm[tid];
#pragma unroll
        for (int o = 4; o; o >>= 1) {
            wx += __shfl_xor(wx, o, 32);
            wy += __shfl_xor(wy, o, 32);
            wm += __shfl_xor(wm, o, 32);
        }
        if (tid == 0) partials[blk] = make_float4(wx, wy, wm, 0.f);
    }
}

// ---------------------------------------------------------------------------
// K2: single block reduces the 2048 block partials -> target point in ws[0].
// ---------------------------------------------------------------------------
__global__ __launch_bounds__(1024) void k2_finalize(
    const float4* __restrict__ partials, float4* __restrict__ target)
{
    const int t = threadIdx.x;
    const float4 a = partials[t];
    const float4 b = partials[t + 1024];
    float wx = a.x + b.x, wy = a.y + b.y, wm = a.z + b.z;
#pragma unroll
    for (int o = 16; o; o >>= 1) {
        wx += __shfl_xor(wx, o, 32);
        wy += __shfl_xor(wy, o, 32);
        wm += __shfl_xor(wm, o, 32);
    }
    __shared__ float sx[32], sy[32], sm[32];
    const int w = t >> 5, l = t & 31;
    if (l == 0) { sx[w] = wx; sy[w] = wy; sm[w] = wm; }
    __syncthreads();
    if (w == 0) {
        wx = sx[l]; wy = sy[l]; wm = sm[l];
#pragma unroll
        for (int o = 16; o; o >>= 1) {
            wx += __shfl_xor(wx, o, 32);
            wy += __shfl_xor(wy, o, 32);
            wm += __shfl_xor(wm, o, 32);
        }
        if (l == 0) {
            const float inv = 1.0f / wm;
            *target = make_float4(wx * inv, wy * inv, wm, 0.f);
        }
    }
}

// ---------------------------------------------------------------------------
// K3: recompute abs_pos per endpoint and scatter out[idx] = vel[idx]+impulse
//     (plain stores: last-writer-wins matches jnp .set duplicate semantics).
//     K1 warmed L2 (working set ~160MB vs 192MB global L2), so these random
//     gathers are mostly L2 hits.
// ---------------------------------------------------------------------------
__global__ __launch_bounds__(TPB) void k3_scatter(
    const int* __restrict__ from_b, const int* __restrict__ to_b,
    const float2* __restrict__ from_rp, const float2* __restrict__ to_rp,
    const float2* __restrict__ pos, const float* __restrict__ ang,
    const float* __restrict__ stf, const float2* __restrict__ vel2,
    const float4* __restrict__ target, float2* __restrict__ out2)
{
    const int tid = threadIdx.x;
    const int e0  = blockIdx.x * (TPB * EPT);
    const int* __restrict__ src =
        (e0 < NCONST) ? (from_b + e0) : (to_b + (e0 - NCONST));

    const int4 ia = *(const int4*)(src + tid * 8);
    const int4 ib = *(const int4*)(src + tid * 8 + 4);
    const float4 tgt   = *target;          // uniform, L2-resident
    const float  scale = stf[0] * DT_F;

    const int idxs[8] = {ia.x, ia.y, ia.z, ia.w, ib.x, ib.y, ib.z, ib.w};
#pragma unroll
    for (int i = 0; i < 8; ++i) {
        const int idx = idxs[i];
        const float2 p = pos[idx];
        const float  a = ang[idx];
        const float2* rp = (idx < NCONST) ? from_rp : to_rp;
        const float2 r = rp[idx & (NCONST - 1)];
        float sa, ca;
        __sincosf(a, &sa, &ca);
        const float c = sa, s = -ca;
        const float ax = fmaf(c, r.x, fmaf(-s, r.y, p.x));
        const float ay = fmaf(s, r.x, fmaf( c, r.y, p.y));
        const float2 v = vel2[idx];
        out2[idx] = make_float2(fmaf(scale, tgt.x - ax, v.x),
                                fmaf(scale, tgt.y - ay, v.y));
    }
}

// ---------------------------------------------------------------------------
extern "C" void kernel_launch(void* const* d_in, const int* in_sizes, int n_in,
                              void* d_out, int out_size, void* d_ws, size_t ws_size,
                              hipStream_t stream) {
    const int*    from_b  = (const int*)d_in[0];
    const int*    to_b    = (const int*)d_in[1];
    const float2* from_rp = (const float2*)d_in[2];
    const float2* to_rp   = (const float2*)d_in[3];
    const float*  stf     = (const float*)d_in[4];
    const float2* pos     = (const float2*)d_in[5];
    const float*  ang     = (const float*)d_in[6];
    const float*  mass    = (const float*)d_in[7];
    const float*  vel     = (const float*)d_in[8];

    float4* ws4      = (float4*)d_ws;
    float4* target   = ws4;        // 16 B
    float4* partials = ws4 + 1;    // NBLK * 16 B = 32 KB

    k1_reduce_copy<<<NBLK, TPB, 0, stream>>>(
        from_b, to_b, from_rp, to_rp, pos, ang, mass,
        (const float4*)vel, (float4*)d_out, partials);
    k2_finalize<<<1, 1024, 0, stream>>>(partials, target);
    k3_scatter<<<NBLK, TPB, 0, stream>>>(
        from_b, to_b, from_rp, to_rp, pos, ang, stf,
        (const float2*)vel, target, (float2*)d_out);
}